// IntLayerNormOptimized_34668976013536
// MI455X (gfx1250) — compile-verified
//
#include <hip/hip_runtime.h>

#define D_DIM 1024
#define WAVES_PER_BLOCK 8

// ---------- CDNA5 async global->LDS helpers ----------

typedef int v4i __attribute__((__vector_size__(16)));
typedef __attribute__((address_space(1))) v4i* global_v4i_ptr;
typedef __attribute__((address_space(3))) v4i* shared_v4i_ptr;

__device__ __forceinline__ void async_load_row(const float* gsrc, float* lds_dst, int lane) {
#if __has_builtin(__builtin_amdgcn_global_load_async_to_lds_b128)
#pragma unroll
  for (int i = 0; i < 8; ++i) {
    __builtin_amdgcn_global_load_async_to_lds_b128(
        (global_v4i_ptr)(gsrc + i * 128 + lane * 4),
        (shared_v4i_ptr)(lds_dst + i * 128 + lane * 4),
        0, 0);
  }
#else
#pragma unroll
  for (int i = 0; i < 8; ++i) {
    auto* lp = (__attribute__((address_space(3))) float*)(lds_dst + i * 128 + lane * 4);
    auto* gp = (__attribute__((address_space(1))) const float*)(gsrc + i * 128 + lane * 4);
    asm volatile("global_load_async_to_lds_b128 %0, %1, off" ::"v"(lp), "v"(gp) : "memory");
  }
#endif
}

template <int N>
__device__ __forceinline__ void wait_asynccnt() {
  asm volatile("s_wait_asynccnt %0" ::"n"(N) : "memory");
}

// ---------- kernel ----------

__global__ __launch_bounds__(256) void int_layernorm_kernel(
    const float* __restrict__ x, const float* __restrict__ weight,
    const float* __restrict__ bias, float* __restrict__ out, int nrows) {
  __shared__ __align__(16) float xbuf[2][WAVES_PER_BLOCK][D_DIM];  // 64 KB
  __shared__ __align__(16) int wq[D_DIM];                          // 4 KB
  __shared__ __align__(16) int bq[D_DIM];                          // 4 KB

  // Quantize weight/bias once per block (Q8, round-half-even like jnp.round).
  for (int i = threadIdx.x; i < D_DIM; i += blockDim.x) {
    wq[i] = (int)rintf(weight[i] * 256.0f);
    bq[i] = (int)rintf(bias[i] * 256.0f);
  }
  __syncthreads();

  const int wave = threadIdx.x >> 5;
  const int lane = threadIdx.x & 31;
  const int waveStride = gridDim.x * WAVES_PER_BLOCK;

  int row = blockIdx.x * WAVES_PER_BLOCK + wave;
  int buf = 0;
  if (row < nrows) async_load_row(x + (size_t)row * D_DIM, &xbuf[0][wave][0], lane);

  for (; row < nrows; row += waveStride) {
    const int nextRow = row + waveStride;
    const bool pre = (nextRow < nrows);
    if (pre) async_load_row(x + (size_t)nextRow * D_DIM, &xbuf[buf ^ 1][wave][0], lane);
    // Wait for the current row's 8 async transfers (completion is in-order).
    if (pre) wait_asynccnt<8>(); else wait_asynccnt<0>();

    const float* rowbuf = &xbuf[buf][wave][0];

    // Pass 1: round to int, row sum.
    int xi[32];
    int sum = 0;
#pragma unroll
    for (int i = 0; i < 8; ++i) {
      float4 v = *(const float4*)(rowbuf + i * 128 + lane * 4);
      int a = (int)rintf(v.x), b2 = (int)rintf(v.y);
      int c = (int)rintf(v.z), d = (int)rintf(v.w);
      xi[4 * i + 0] = a; xi[4 * i + 1] = b2; xi[4 * i + 2] = c; xi[4 * i + 3] = d;
      sum += a + b2 + c + d;
    }
#pragma unroll
    for (int off = 16; off; off >>= 1) sum += __shfl_xor(sum, off, 32);

    const int mean = (sum * 64) >> 16;  // (sum_x * INV_D) >> Q_D, exact in int32

    // Pass 2: center, sum of squares.
    int sumsq = 0;
#pragma unroll
    for (int e = 0; e < 32; ++e) {
      int y = xi[e] - mean;
      xi[e] = y;
      sumsq += y * y;
    }
#pragma unroll
    for (int off = 16; off; off >>= 1) sumsq += __shfl_xor(sumsq, off, 32);

    int var = sumsq >> 10;  // (sum_sq*64)>>16, sumsq >= 0 so exact
    if (var < 1) var = 1;

    // k = floor(log2(var)) exactly; matches reference f32-log2 floor in range.
    const int k = 31 - __builtin_clz((unsigned)var);
    const int parity = k & 1;
    const int sa = k - 4;
    const int mant = (sa >= 0) ? ((var >> sa) & 15) : ((var << (-sa)) & 15);
    // Recompute LUT entry exactly as make_isqrt_lut (double, round-half-even).
    const double lv = (double)(1 << parity) * (1.0 + (double)mant * 0.0625);
    const int inv = (int)__builtin_llrint(32768.0 / sqrt(lv));
    const int tshift = (k >> 1) + 15;  // p + SHIFT + Q_LUT

    float* orow = out + (size_t)row * D_DIM;
#pragma unroll
    for (int i = 0; i < 8; ++i) {
      const int col = i * 128 + lane * 4;
      int4 wv = *(const int4*)(wq + col);
      int4 bv = *(const int4*)(bq + col);
      float4 o;
      // y*inv fits int32 (|y|<~600, inv<=32768); widen before *w (can exceed 2^31).
      o.x = (float)((int)(((long long)(xi[4 * i + 0] * inv) * wv.x) >> tshift) + bv.x) * 0.00390625f;
      o.y = (float)((int)(((long long)(xi[4 * i + 1] * inv) * wv.y) >> tshift) + bv.y) * 0.00390625f;
      o.z = (float)((int)(((long long)(xi[4 * i + 2] * inv) * wv.z) >> tshift) + bv.z) * 0.00390625f;
      o.w = (float)((int)(((long long)(xi[4 * i + 3] * inv) * wv.w) >> tshift) + bv.w) * 0.00390625f;
      *(float4*)(orow + col) = o;
    }

    buf ^= 1;
  }
}

// ---------- launch ----------

extern "C" void kernel_launch(void* const* d_in, const int* in_sizes, int n_in,
                              void* d_out, int out_size, void* d_ws, size_t ws_size,
                              hipStream_t stream) {
  (void)n_in; (void)d_ws; (void)ws_size; (void)out_size;
  const float* x = (const float*)d_in[0];
  const float* w = (const float*)d_in[1];
  const float* b = (const float*)d_in[2];
  // d_in[3] (isqrt_lut) is intentionally unused: the 32-entry table is a pure
  // function of module constants and is recomputed exactly on-device.
  float* out = (float*)d_out;

  int nrows = in_sizes[0] / D_DIM;
  int blocks = (nrows + WAVES_PER_BLOCK - 1) / WAVES_PER_BLOCK;
  if (blocks > 512) blocks = 512;  // ~4 rows/wave -> double-buffer pays off
  if (blocks < 1) blocks = 1;

  hipLaunchKernelGGL(int_layernorm_kernel, dim3(blocks), dim3(256), 0, stream,
                     x, w, b, out, nrows);
}